// SimpleConv2d_12910671691816
// MI455X (gfx1250) — compile-verified
//
#include <hip/hip_runtime.h>

// ---------------------------------------------------------------------------
// Implicit-GEMM 3x3 conv (stride 1, pad 1) on MI455X / gfx1250 (wave32).
//   M = out-channels (256), N = batch*H*W (100352), K' = rs*128 + c (1152)
// K reordered so each BK=16 tile has uniform (r,s): im2col decode is scalar.
// 64(ch) x 128(sp) block tile, 8 waves, 32x32 per wave (4 WMMA accumulators),
// LDS double-buffered, v2f (b64) fragment loads, V_WMMA_F32_16X16X4_F32.
// ---------------------------------------------------------------------------

typedef float v2f __attribute__((ext_vector_type(2)));
typedef float v8f __attribute__((ext_vector_type(8)));

#define CIN   128
#define KOUT  256
#define HIMG  56
#define WIMG  56
#define HW    3136           // 56*56
#define KTOT  1152           // 9*128
#define MTOT  100352         // 32*3136 (GEMM N)
#define BCH   64             // channel tile (GEMM M)
#define BSP   128            // spatial tile (GEMM N)
#define BK    16             // K tile -> 4 WMMA k-steps
#define LDK   20             // padded LDS row stride (even: keeps v2f aligned)
#define NITER (KTOT / BK)    // 72

struct Regs {
    float a[4];   // weight tile stage: rows (am, am+32) x k (ak, ak+1)
    float b[8];   // x tile stage:      row bn x k (bk0 .. bk0+7)
};

__global__ __launch_bounds__(256) void conv3x3_wmma_f32(
    const float* __restrict__ x,      // (32,128,56,56)
    const float* __restrict__ wgt,    // (256,128,3,3)
    const float* __restrict__ bias,   // (256,)
    float* __restrict__ out)          // (32,256,56,56)
{
    __shared__ float As[2][BCH][LDK];  // [buf][channel][k]
    __shared__ float Bs[2][BSP][LDK];  // [buf][spatial][k]

    const int tid  = threadIdx.x;
    const int lane = tid & 31;
    const int wave = tid >> 5;         // 0..7
    const int wm   = wave & 1;         // channel half (32 ch)
    const int wn   = wave >> 1;        // spatial quarter (32 sp)
    const int hi   = lane >> 4;        // half-wave: K pair select / M+8
    const int l16  = lane & 15;

    const int chBase = blockIdx.y * BCH;   // 0,64,128,192
    const int spBase = blockIdx.x * BSP;

    // ---- B (im2col x) load indices: 128 sp x 16 k, 8 elems/thread ----
    const int bn  = tid & 127;             // spatial col in tile
    const int bk0 = (tid >> 7) * 8;        // k = bk0 .. bk0+7
    const int spb = spBase + bn;
    const int bbB = spb / HW;
    const int hwB = spb - bbB * HW;
    const int ohB = hwB / WIMG;
    const int owB = hwB - ohB * WIMG;
    const float* xb = x + (long)bbB * CIN * HW + hwB;  // + c*HW + roff

    // ---- A (weight) load indices: 64 ch x 16 k, 4 elems/thread ----
    const int am = tid >> 3;               // rows am, am+32
    const int ak = (tid & 7) * 2;          // k pair
    const float* wp0 = wgt + (long)(chBase + am) * KTOT + ak * 9;
    const float* wp1 = wp0 + 32 * KTOT;

    v8f acc[2][2] = {{{}, {}}, {{}, {}}};  // [ch sub 16][sp sub 16]

    auto gload = [&](int k0, Regs& rg) {
        const int rs = k0 >> 7;            // uniform within tile (16 | 128)
        const int c0 = k0 & 127;
        const int r  = rs / 3;             // scalar constant-div
        const int s  = rs - r * 3;
        const int roff = (r - 1) * WIMG + (s - 1);
        const bool vld = ((unsigned)(ohB + r - 1) < (unsigned)HIMG) &&
                         ((unsigned)(owB + s - 1) < (unsigned)WIMG);
#pragma unroll
        for (int j = 0; j < 8; ++j) rg.b[j] = 0.0f;
        if (vld) {
            const float* p = xb + (long)(c0 + bk0) * HW + roff;
#pragma unroll
            for (int j = 0; j < 8; ++j) rg.b[j] = p[(long)j * HW];
        }
        const long wo = (long)c0 * 9 + rs;
        rg.a[0] = wp0[wo];
        rg.a[1] = wp0[wo + 9];
        rg.a[2] = wp1[wo];
        rg.a[3] = wp1[wo + 9];
    };

    auto lstore = [&](int buf, const Regs& rg) {
#pragma unroll
        for (int j = 0; j < 4; ++j) {
            v2f t = {rg.b[2 * j], rg.b[2 * j + 1]};
            *(v2f*)&Bs[buf][bn][bk0 + 2 * j] = t;
        }
        v2f ta0 = {rg.a[0], rg.a[1]};
        v2f ta1 = {rg.a[2], rg.a[3]};
        *(v2f*)&As[buf][am][ak]      = ta0;
        *(v2f*)&As[buf][am + 32][ak] = ta1;
    };

    auto compute = [&](int buf) {
#pragma unroll
        for (int kk = 0; kk < BK; kk += 4) {
            const int krow = kk + hi * 2;  // lanes0-15: K{0,1}; 16-31: K{2,3}
            v2f a0 = *(const v2f*)&As[buf][wm * 32 + l16][krow];
            v2f a1 = *(const v2f*)&As[buf][wm * 32 + 16 + l16][krow];
            v2f b0 = *(const v2f*)&Bs[buf][wn * 32 + l16][krow];
            v2f b1 = *(const v2f*)&Bs[buf][wn * 32 + 16 + l16][krow];
            acc[0][0] = __builtin_amdgcn_wmma_f32_16x16x4_f32(
                            false, a0, false, b0, (short)0, acc[0][0], false, false);
            acc[0][1] = __builtin_amdgcn_wmma_f32_16x16x4_f32(
                            false, a0, false, b1, (short)0, acc[0][1], false, false);
            acc[1][0] = __builtin_amdgcn_wmma_f32_16x16x4_f32(
                            false, a1, false, b0, (short)0, acc[1][0], false, false);
            acc[1][1] = __builtin_amdgcn_wmma_f32_16x16x4_f32(
                            false, a1, false, b1, (short)0, acc[1][1], false, false);
        }
    };

    // ---- prologue: stage tile 0 ----
    Regs rg;
    gload(0, rg);
    lstore(0, rg);
    __syncthreads();

    // ---- main loop: double-buffered, one barrier per K tile ----
    int cur = 0;
#pragma unroll 1
    for (int it = 0; it < NITER - 1; ++it) {
        gload((it + 1) * BK, rg);     // global loads overlap WMMAs below
        compute(cur);
        lstore(cur ^ 1, rg);
        __syncthreads();
        cur ^= 1;
    }
    compute(cur);

    // ---- epilogue: lanes = consecutive hw -> coalesced stores ----
#pragma unroll
    for (int i = 0; i < 2; ++i) {
#pragma unroll
        for (int j = 0; j < 2; ++j) {
            const int sp0 = spBase + wn * 32 + j * 16 + l16;
            const int b2  = sp0 / HW;
            const int hw2 = sp0 - b2 * HW;
            float* op = out + (long)b2 * KOUT * HW + hw2;
#pragma unroll
            for (int rr = 0; rr < 8; ++rr) {
                const int ch = chBase + wm * 32 + i * 16 + hi * 8 + rr;
                op[(long)ch * HW] = acc[i][j][rr] + bias[ch];
            }
        }
    }
}

extern "C" void kernel_launch(void* const* d_in, const int* in_sizes, int n_in,
                              void* d_out, int out_size, void* d_ws, size_t ws_size,
                              hipStream_t stream) {
    const float* x    = (const float*)d_in[0];
    const float* wgt  = (const float*)d_in[1];
    const float* bias = (const float*)d_in[2];
    float* out        = (float*)d_out;

    dim3 grid(MTOT / BSP, KOUT / BCH);   // (784, 4)
    conv3x3_wmma_f32<<<grid, 256, 0, stream>>>(x, wgt, bias, out);
}